// MicrotensorLayerNorm_1872605741567
// MI455X (gfx1250) — compile-verified
//
#include <hip/hip_runtime.h>

typedef __attribute__((ext_vector_type(4))) float f4;
typedef __attribute__((ext_vector_type(4))) int   v4i;

#define FEATURES 1024
#define NCHUNK (FEATURES / (32 * 4))   // 8 float4 chunks per lane per row
#define LN_EPS 1e-5f

__global__ __launch_bounds__(256) void ln_fwd_kernel(
    const float* __restrict__ x,
    const float* __restrict__ alpha,
    const float* __restrict__ bias,
    float* __restrict__ out,
    int nrows)
{
    __shared__ f4 s_alpha[FEATURES / 4];   // 4 KB
    __shared__ f4 s_bias[FEATURES / 4];    // 4 KB

    const int t = threadIdx.x;

    // ---- Stage alpha/bias into LDS once per block (256 threads x 16B = 4KB each)
#if defined(__gfx1250__) && __has_builtin(__builtin_amdgcn_global_load_async_to_lds_b128) && __has_builtin(__builtin_amdgcn_s_wait_asynccnt)
    {
        typedef __attribute__((address_space(1))) v4i* gp_t;  // global (AS1) int4*
        typedef __attribute__((address_space(3))) v4i* lp_t;  // LDS (AS3) int4*
        gp_t ga = (gp_t)(alpha + 4 * t);
        gp_t gb = (gp_t)(bias  + 4 * t);
        lp_t la = (lp_t)(&s_alpha[t]);
        lp_t lb = (lp_t)(&s_bias[t]);
        __builtin_amdgcn_global_load_async_to_lds_b128(ga, la, 0, 0);
        __builtin_amdgcn_global_load_async_to_lds_b128(gb, lb, 0, 0);
        __builtin_amdgcn_s_wait_asynccnt(0);
    }
#else
    s_alpha[t] = ((const f4*)alpha)[t];
    s_bias[t]  = ((const f4*)bias)[t];
#endif
    __syncthreads();

    const int lane = t & 31;
    const int wave = t >> 5;
    const int wpb  = blockDim.x >> 5;
    const int wave_id = blockIdx.x * wpb + wave;
    const int wstride = gridDim.x * wpb;

    for (int row = wave_id; row < nrows; row += wstride) {
        const f4* __restrict__ xr   = (const f4*)(x   + (size_t)row * FEATURES);
        f4* __restrict__       orow = (f4*)      (out + (size_t)row * FEATURES);

        // ---- Single streaming pass: row lives in 32 VGPRs/lane
        f4 v[NCHUNK];
        float sum = 0.0f, sumsq = 0.0f;
#pragma unroll
        for (int c = 0; c < NCHUNK; ++c) {
            v[c] = __builtin_nontemporal_load(xr + c * 32 + lane);
            sum   += (v[c].x + v[c].y) + (v[c].z + v[c].w);
            sumsq += v[c].x * v[c].x + v[c].y * v[c].y
                   + v[c].z * v[c].z + v[c].w * v[c].w;
        }

        // ---- wave32 butterfly reduction
#pragma unroll
        for (int m = 16; m >= 1; m >>= 1) {
            sum   += __shfl_xor(sum,   m, 32);
            sumsq += __shfl_xor(sumsq, m, 32);
        }

        const float inv_n = 1.0f / (float)FEATURES;
        const float mean  = sum * inv_n;
        const float var   = fmaf(-mean, mean, sumsq * inv_n);
        const float rstd  = rsqrtf(var + LN_EPS);

        // ---- normalize + affine, NT store
#pragma unroll
        for (int c = 0; c < NCHUNK; ++c) {
            const int idx = c * 32 + lane;
            f4 a = s_alpha[idx];
            f4 b = s_bias[idx];
            f4 r;
            r.x = fmaf((v[c].x - mean) * rstd, a.x, b.x);
            r.y = fmaf((v[c].y - mean) * rstd, a.y, b.y);
            r.z = fmaf((v[c].z - mean) * rstd, a.z, b.z);
            r.w = fmaf((v[c].w - mean) * rstd, a.w, b.w);
            __builtin_nontemporal_store(r, orow + idx);
        }
    }
}

extern "C" void kernel_launch(void* const* d_in, const int* in_sizes, int n_in,
                              void* d_out, int out_size, void* d_ws, size_t ws_size,
                              hipStream_t stream) {
    const float* x     = (const float*)d_in[0];
    const float* alpha = (const float*)d_in[1];
    const float* bias  = (const float*)d_in[2];
    float*       out   = (float*)d_out;

    const int nrows = in_sizes[0] / FEATURES;   // 4 * 8192 = 32768
    const int threads = 256;                    // 8 wave32 per block
    const int wpb = threads / 32;
    int blocks = (nrows + wpb - 1) / wpb;
    if (blocks > 4096) blocks = 4096;           // 4096*8 waves = 1 row/wave here
    if (blocks < 1) blocks = 1;

    hipLaunchKernelGGL(ln_fwd_kernel, dim3(blocks), dim3(threads), 0, stream,
                       x, alpha, bias, out, nrows);
}